// GlobalGraphGAT_43757126812043
// MI455X (gfx1250) — compile-verified
//
#include <hip/hip_runtime.h>
#include <cstdint>

#define NEG_SLOPE 0.2f

typedef __attribute__((ext_vector_type(2))) float v2f;
typedef __attribute__((ext_vector_type(8))) float v8f;

__device__ __forceinline__ float lrelu(float x) { return x > 0.f ? x : NEG_SLOPE * x; }

// Monotone order-preserving float<->uint encoding so float segment-max can use
// integer atomicMax. enc(f) of any real float is > 0, so zero-init == -inf.
__device__ __forceinline__ unsigned enc_f32(float f) {
    unsigned u = __float_as_uint(f);
    return (u & 0x80000000u) ? ~u : (u | 0x80000000u);
}
__device__ __forceinline__ float dec_f32(unsigned u) {
    return (u & 0x80000000u) ? __uint_as_float(u ^ 0x80000000u) : __uint_as_float(~u);
}

__global__ void fill_zero_u32(uint32_t* __restrict__ p, long n) {
    long i = (long)blockIdx.x * blockDim.x + threadIdx.x;
    if (i < n) p[i] = 0u;
}

// ---------------------------------------------------------------------------
// C[M,NCOLS] = A[M,K] @ B[K,NCOLS] in fp32 via V_WMMA_F32_16X16X4_F32.
// K/NCOLS are template constants: all per-k addresses are immediate offsets
// from two hoisted base pointers (no per-iteration address math). Each wave
// computes a 16x64 strip (4 accumulators) reusing the A fragment 4x; all four
// B fragments are fetched before the WMMA group so the scheduler can overlap
// loads with matrix issue (partial loadcnt waits instead of 0x0 per WMMA).
// grid = (ceil(M/64), NCOLS/64), block = 128 (4 waves, one 16-row tile each).
// A frag (16x4 fp32): lanes 0-15 rows M=0..15; VGPR0 = K(0|2), VGPR1 = K(1|3)
// split by lane half. C/D: VGPR r holds M = r + 8*(lane>>4), N = lane&15.
// ---------------------------------------------------------------------------
template <int K, int NCOLS>
__global__ void gemm_wmma_f32(const float* __restrict__ A, const float* __restrict__ B,
                              float* __restrict__ C, int M) {
    const int lane  = threadIdx.x & 31;
    const int wave  = threadIdx.x >> 5;
    const int halfk = lane >> 4;          // 0 or 1
    const int l15   = lane & 15;
    const int tileM = blockIdx.x * 4 + wave;   // 16-row tile index
    if (tileM * 16 >= M) return;               // whole idle wave exits
    int row = tileM * 16 + l15;
    if (row >= M) row = M - 1;                 // clamp (stores predicated below)
    const int colbase = blockIdx.y * 64;

    v8f acc[4];
#pragma unroll
    for (int nt = 0; nt < 4; ++nt) acc[nt] = (v8f){0.f,0.f,0.f,0.f,0.f,0.f,0.f,0.f};

    // Hoisted lane-dependent bases; everything below uses immediate offsets.
    const float* abase = A + (size_t)row * K + 2 * halfk;
    const float* bbase = B + (size_t)colbase + l15 + (size_t)(2 * halfk) * NCOLS;

#pragma unroll
    for (int k0 = 0; k0 < K; k0 += 4) {
        v2f a, b[4];
        a.x = abase[k0];
        a.y = abase[k0 + 1];
#pragma unroll
        for (int nt = 0; nt < 4; ++nt) {
            b[nt].x = bbase[k0 * NCOLS + nt * 16];
            b[nt].y = bbase[(k0 + 1) * NCOLS + nt * 16];
        }
#pragma unroll
        for (int nt = 0; nt < 4; ++nt)
            acc[nt] = __builtin_amdgcn_wmma_f32_16x16x4_f32(false, a, false, b[nt],
                                                            (short)0, acc[nt], false, false);
    }
    const int rbase = tileM * 16 + 8 * halfk;
#pragma unroll
    for (int nt = 0; nt < 4; ++nt) {
#pragma unroll
        for (int r = 0; r < 8; ++r) {
            const int orow = rbase + r;
            if (orow < M)
                C[(size_t)orow * NCOLS + colbase + nt * 16 + l15] = acc[nt][r];
        }
    }
}

// ---------------------------------------------------------------------------
// Per-node attention scores: s_src[n,h] = sum_c xp[n,h,c]*att_src[h,c] (+dst).
// One wave per node, coalesced row reads, shuffle reduction.
// ---------------------------------------------------------------------------
template <int H, int CH>
__global__ void attn_scores(const float* __restrict__ xp, const float* __restrict__ att_src,
                            const float* __restrict__ att_dst, float* __restrict__ s_src,
                            float* __restrict__ s_dst, int n_nodes) {
    const int wid  = (int)(((long)blockIdx.x * blockDim.x + threadIdx.x) >> 5);
    const int lane = threadIdx.x & 31;
    if (wid >= n_nodes) return;
    constexpr int W = H * CH;             // row width (CH multiple of 32)
    float as[H], ad[H];
#pragma unroll
    for (int h = 0; h < H; ++h) { as[h] = 0.f; ad[h] = 0.f; }
    const float* row = xp + (size_t)wid * W;
#pragma unroll
    for (int j = 0; j < W / 32; ++j) {
        const int c = j * 32 + lane;
        const int h = (j * 32) / CH;      // uniform per iteration
        const float v = row[c];
        as[h] += v * att_src[c];
        ad[h] += v * att_dst[c];
    }
#pragma unroll
    for (int off = 16; off > 0; off >>= 1) {
#pragma unroll
        for (int h = 0; h < H; ++h) {
            as[h] += __shfl_xor(as[h], off, 32);
            ad[h] += __shfl_xor(ad[h], off, 32);
        }
    }
    if (lane == 0) {
#pragma unroll
        for (int h = 0; h < H; ++h) {
            s_src[(size_t)wid * H + h] = as[h];
            s_dst[(size_t)wid * H + h] = ad[h];
        }
    }
}

__device__ __forceinline__ void edge_endpoints(const int* __restrict__ srcs,
                                               const int* __restrict__ dsts,
                                               int e, int E, int& s, int& d) {
    if (e < E) { s = srcs[e]; d = dsts[e]; }
    else       { s = e - E;  d = e - E;  }  // appended self loops
}

template <int H>
__global__ void edge_max(const int* __restrict__ srcs, const int* __restrict__ dsts,
                         int E, int E2, const float* __restrict__ s_src,
                         const float* __restrict__ s_dst, unsigned* __restrict__ maxenc) {
    long t = (long)blockIdx.x * blockDim.x + threadIdx.x;
    if (t >= (long)E2 * H) return;
    const int e = (int)(t / H), h = (int)(t % H);
    int s, d;
    edge_endpoints(srcs, dsts, e, E, s, d);
    const float sc = lrelu(s_src[(size_t)s * H + h] + s_dst[(size_t)d * H + h]);
    atomicMax(&maxenc[(size_t)d * H + h], enc_f32(sc));
}

// Computes e = exp(score - max[dst]) once, caches it per (edge,head) in
// `ealpha` (unique index -> plain store) and atomically accumulates denom.
template <int H>
__global__ void edge_expsum(const int* __restrict__ srcs, const int* __restrict__ dsts,
                            int E, int E2, const float* __restrict__ s_src,
                            const float* __restrict__ s_dst,
                            const unsigned* __restrict__ maxenc, float* __restrict__ denom,
                            float* __restrict__ ealpha) {
    long t = (long)blockIdx.x * blockDim.x + threadIdx.x;
    if (t >= (long)E2 * H) return;
    const int e = (int)(t / H), h = (int)(t % H);
    int s, d;
    edge_endpoints(srcs, dsts, e, E, s, d);
    const float sc = lrelu(s_src[(size_t)s * H + h] + s_dst[(size_t)d * H + h]);
    const float ex = __expf(sc - dec_f32(maxenc[(size_t)d * H + h]));
    ealpha[t] = ex;
    atomicAdd(&denom[(size_t)d * H + h], ex);
}

// One wave per edge; src/dst are wave-uniform so force them into SGPRs with
// readfirstlane (scalar address math, broadcast alpha). Lanes sweep channels
// in coalesced 128B slices; atomics resolve in the 192MB L2.
template <int H, int CH>
__global__ void edge_aggregate(const int* __restrict__ srcs, const int* __restrict__ dsts,
                               int E, int E2, const float* __restrict__ ealpha,
                               const float* __restrict__ denom,
                               const float* __restrict__ xp, float* __restrict__ agg) {
    const long t   = (long)blockIdx.x * blockDim.x + threadIdx.x;
    const int e    = (int)(t >> 5);              // uniform across the wave
    const int lane = threadIdx.x & 31;
    if (e >= E2) return;
    int s, d;
    edge_endpoints(srcs, dsts, e, E, s, d);
    s = __builtin_amdgcn_readfirstlane(s);
    d = __builtin_amdgcn_readfirstlane(d);
    float alpha[H];
#pragma unroll
    for (int h = 0; h < H; ++h)
        alpha[h] = ealpha[(size_t)e * H + h] / (denom[(size_t)d * H + h] + 1e-16f);
    const float* xrow = xp  + (size_t)s * (H * CH);
    float*       orow = agg + (size_t)d * (H * CH);
#pragma unroll
    for (int j = 0; j < (H * CH) / 32; ++j) {
        const int c = j * 32 + lane;
        atomicAdd(&orow[c], xrow[c] * alpha[(j * 32) / CH]);
    }
}

template <bool RELU>
__global__ void bias_act(float* __restrict__ buf, const float* __restrict__ bias,
                         long n, int width) {
    long i = (long)blockIdx.x * blockDim.x + threadIdx.x;
    if (i >= n) return;
    float v = buf[i] + bias[i % width];
    buf[i] = RELU ? fmaxf(v, 0.f) : v;
}

static inline unsigned nblocks(long total, int bs) {
    return (unsigned)((total + bs - 1) / bs);
}

extern "C" void kernel_launch(void* const* d_in, const int* in_sizes, int n_in,
                              void* d_out, int out_size, void* d_ws, size_t ws_size,
                              hipStream_t stream) {
    (void)n_in; (void)out_size; (void)ws_size;
    const float* x        = (const float*)d_in[0];
    const int*   ei       = (const int*)d_in[1];
    const float* W1       = (const float*)d_in[2];
    const float* att_src1 = (const float*)d_in[3];
    const float* att_dst1 = (const float*)d_in[4];
    const float* bias1    = (const float*)d_in[5];
    const float* W2       = (const float*)d_in[6];
    const float* att_src2 = (const float*)d_in[7];
    const float* att_dst2 = (const float*)d_in[8];
    const float* bias2    = (const float*)d_in[9];

    const int n  = in_sizes[0] / 128;   // 50000 nodes
    const int E  = in_sizes[1] / 2;     // 800000 edges
    const int E2 = E + n;               // + self loops
    const int* srcs = ei;
    const int* dsts = ei + E;

    // Workspace carve-up (~135 MB, hot set L2-resident).
    char* w = (char*)d_ws;
    float*    xp1    = (float*)w;    w += (size_t)n * 256 * 4;  // layer1 features
    float*    agg1   = (float*)w;    w += (size_t)n * 256 * 4;  // layer1 agg -> h (in place)
    float*    xp2    = (float*)w;    w += (size_t)n * 64 * 4;   // layer2 features
    float*    ealpha = (float*)w;    w += (size_t)E2 * 4 * 4;   // cached exp() per edge-head
    float*    ssrc1  = (float*)w;    w += (size_t)n * 4 * 4;
    float*    sdst1  = (float*)w;    w += (size_t)n * 4 * 4;
    unsigned* max1   = (unsigned*)w; w += (size_t)n * 4 * 4;
    float*    den1   = (float*)w;    w += (size_t)n * 4 * 4;
    float*    ssrc2  = (float*)w;    w += (size_t)n * 4;
    float*    sdst2  = (float*)w;    w += (size_t)n * 4;
    unsigned* max2   = (unsigned*)w; w += (size_t)n * 4;
    float*    den2   = (float*)w;    w += (size_t)n * 4;
    float*    out    = (float*)d_out;

    const int BS = 256;
    const unsigned gx = (unsigned)((n + 63) / 64);   // GEMM row-tile blocks

    // ---------------- Layer 1 (4 heads x 64 ch, concat) ----------------
    gemm_wmma_f32<128, 256><<<dim3(gx, 4), 128, 0, stream>>>(x, W1, xp1, n);
    attn_scores<4, 64><<<nblocks((long)n * 32, BS), BS, 0, stream>>>(xp1, att_src1, att_dst1,
                                                                     ssrc1, sdst1, n);
    fill_zero_u32<<<nblocks((long)n * 4, BS), BS, 0, stream>>>(max1, (long)n * 4);
    fill_zero_u32<<<nblocks((long)n * 4, BS), BS, 0, stream>>>((uint32_t*)den1, (long)n * 4);
    fill_zero_u32<<<nblocks((long)n * 256, BS), BS, 0, stream>>>((uint32_t*)agg1, (long)n * 256);
    edge_max<4><<<nblocks((long)E2 * 4, BS), BS, 0, stream>>>(srcs, dsts, E, E2, ssrc1, sdst1, max1);
    edge_expsum<4><<<nblocks((long)E2 * 4, BS), BS, 0, stream>>>(srcs, dsts, E, E2, ssrc1, sdst1,
                                                                 max1, den1, ealpha);
    edge_aggregate<4, 64><<<nblocks((long)E2 * 32, BS), BS, 0, stream>>>(srcs, dsts, E, E2, ealpha,
                                                                         den1, xp1, agg1);
    bias_act<true><<<nblocks((long)n * 256, BS), BS, 0, stream>>>(agg1, bias1, (long)n * 256, 256);

    // ---------------- Layer 2 (1 head x 64 ch, mean==identity) ----------------
    gemm_wmma_f32<256, 64><<<dim3(gx, 1), 128, 0, stream>>>(agg1, W2, xp2, n);
    attn_scores<1, 64><<<nblocks((long)n * 32, BS), BS, 0, stream>>>(xp2, att_src2, att_dst2,
                                                                     ssrc2, sdst2, n);
    fill_zero_u32<<<nblocks((long)n, BS), BS, 0, stream>>>(max2, (long)n);
    fill_zero_u32<<<nblocks((long)n, BS), BS, 0, stream>>>((uint32_t*)den2, (long)n);
    fill_zero_u32<<<nblocks((long)n * 64, BS), BS, 0, stream>>>((uint32_t*)out, (long)n * 64);
    edge_max<1><<<nblocks((long)E2, BS), BS, 0, stream>>>(srcs, dsts, E, E2, ssrc2, sdst2, max2);
    edge_expsum<1><<<nblocks((long)E2, BS), BS, 0, stream>>>(srcs, dsts, E, E2, ssrc2, sdst2,
                                                             max2, den2, ealpha);
    edge_aggregate<1, 64><<<nblocks((long)E2 * 32, BS), BS, 0, stream>>>(srcs, dsts, E, E2, ealpha,
                                                                         den2, xp2, out);
    bias_act<false><<<nblocks((long)n * 64, BS), BS, 0, stream>>>(out, bias2, (long)n * 64, 64);
}